// GraphConv_dense_3676492005535
// MI455X (gfx1250) — compile-verified
//
#include <hip/hip_runtime.h>

// Problem constants (from reference)
#define NN   20000      // nodes
#define DD   32         // input dim
#define BB   4          // batch
#define FF   128        // D*B feature width of diffusion states
#define OUTC 64         // output dim
#define MMET 7          // num metrics M = MAX_STEP*S + 1
#define KW   224        // D*M = W's K dimension
#define EE   640000     // edges per support

typedef __attribute__((ext_vector_type(2))) float v2f;
typedef __attribute__((ext_vector_type(8))) float v8f;

// ---------------------------------------------------------------------------
// x0[n, d*4 + b] = inputs[b, n, d]   (permute([1,2,0]).reshape(n,-1))
// ---------------------------------------------------------------------------
__global__ __launch_bounds__(256) void build_x0(const float* __restrict__ in,
                                                float* __restrict__ x0) {
  int idx = blockIdx.x * 256 + threadIdx.x;          // over NN*FF (exact grid)
  int n = idx / FF;
  int f = idx - n * FF;
  int d = f >> 2;
  int b = f & 3;
  x0[idx] = in[((size_t)b * NN + n) * DD + d];
}

// ---------------------------------------------------------------------------
// y = -x   (elementwise, float4): seeds the Chebyshev recurrence 2*A*x1 - x0
// ---------------------------------------------------------------------------
__global__ __launch_bounds__(256) void neg_copy(float* __restrict__ y,
                                                const float* __restrict__ x) {
  int idx = blockIdx.x * 256 + threadIdx.x;          // over NN*FF/4 (exact)
  float4 v = ((const float4*)x)[idx];
  float4 o; o.x = -v.x; o.y = -v.y; o.z = -v.z; o.w = -v.w;
  ((float4*)y)[idx] = o;
}

// ---------------------------------------------------------------------------
// COO scatter SpMM: y[dst,:] += scale * val * x[src,:]
// One wave per edge; 32 lanes x float4 = 128 features. Atomic adds hit
// consecutive addresses (4 cachelines/edge) -> coalesced L2 atomics; x and y
// are ~10MB each and L2-resident on the 192MB L2.
// ---------------------------------------------------------------------------
__global__ __launch_bounds__(256) void spmm_scatter(const float* __restrict__ vals,
                                                    const int*   __restrict__ src,
                                                    const int*   __restrict__ dst,
                                                    const float* __restrict__ x,
                                                    float*       __restrict__ y,
                                                    float scale) {
  int e = blockIdx.x * 8 + (threadIdx.x >> 5);       // 8 waves/block, exact grid
  int lane = threadIdx.x & 31;
  float v = vals[e] * scale;
  size_t soff = (size_t)src[e] * FF + lane * 4u;
  size_t doff = (size_t)dst[e] * FF + lane * 4u;
  float4 xv = *(const float4*)(x + soff);
  atomicAdd(y + doff + 0, v * xv.x);
  atomicAdd(y + doff + 1, v * xv.y);
  atomicAdd(y + doff + 2, v * xv.z);
  atomicAdd(y + doff + 3, v * xv.w);
}

// ---------------------------------------------------------------------------
// WMMA f32 GEMM-accumulate pass for metric m:
//   out[b, n, o] (+)= sum_{d=0..31} X[n, d*4+b] * W[o, d*7+m]   (+bias if !beta)
// Tile = 16(rows of n) x 16(cols of o), K consumed 4 at a time via
// V_WMMA_F32_16X16X4_F32 (8 wmma per tile). One wave per tile, 8 waves/block.
// VGPR layouts per CDNA5 ISA 7.12.2:
//   A 16x4 : lane L -> row r=L%16, holds K = 2g, 2g+1   (g = L/16) in 2 VGPRs
//   B 4x16 : lane L -> col c=L%16, holds K = 2g, 2g+1           in 2 VGPRs
//   C/D    : lane L -> col c=L%16, VGPR v -> row v + 8g
// Grid divides tiles exactly -> no divergence -> EXEC all-1s for WMMA.
// ---------------------------------------------------------------------------
__global__ __launch_bounds__(256) void gemm_acc(const float* __restrict__ X,
                                                const float* __restrict__ W,
                                                const float* __restrict__ bias,
                                                float* __restrict__ out,
                                                int m, int beta) {
  const int lane = threadIdx.x & 31;
  const int tile = blockIdx.x * 8 + (threadIdx.x >> 5);  // 20000 tiles total
  const int ot = tile & 3;                               // o-tile (4)
  const int nt = (tile >> 2) % 1250;                     // n-tile (1250)
  const int b  = (tile >> 2) / 1250;                     // batch (4)
  const int n0 = nt << 4;
  const int o0 = ot << 4;
  const int r  = lane & 15;
  const int g  = lane >> 4;

  const size_t outBase = ((size_t)b * NN + n0 + 8 * g) * OUTC + (o0 + r);
  v8f c;
  if (beta) {
#pragma unroll
    for (int v = 0; v < 8; ++v) c[v] = out[outBase + (size_t)v * OUTC];
  } else {
    float bz = bias[o0 + r];
#pragma unroll
    for (int v = 0; v < 8; ++v) c[v] = bz;
  }

  const float* xrow = X + (size_t)(n0 + r) * FF + b;   // A row for this lane
  const float* wrow = W + (size_t)(o0 + r) * KW + m;   // B col for this lane

#pragma unroll
  for (int kk = 0; kk < 8; ++kk) {
    const int k0 = kk * 4 + 2 * g;
    v2f a, bm;
    a[0]  = xrow[(size_t)(k0    ) * 4];   // X[n0+r, k0  *4 + b]
    a[1]  = xrow[(size_t)(k0 + 1) * 4];
    bm[0] = wrow[(size_t)(k0    ) * 7];   // W[o0+r, k0  *7 + m]
    bm[1] = wrow[(size_t)(k0 + 1) * 7];
    c = __builtin_amdgcn_wmma_f32_16x16x4_f32(false, a, false, bm,
                                              (short)0, c, false, false);
  }

#pragma unroll
  for (int v = 0; v < 8; ++v) out[outBase + (size_t)v * OUTC] = c[v];
}

// ---------------------------------------------------------------------------
// Host orchestration. 3 rotating [NN,FF] buffers in d_ws (30.7 MB total),
// reproducing the reference's aliasing quirk (support 1 starts from B2 = T2
// of support 0). GEMM is fused as 7 accumulate passes right after each state
// is produced.
// ---------------------------------------------------------------------------
extern "C" void kernel_launch(void* const* d_in, const int* in_sizes, int n_in,
                              void* d_out, int out_size, void* d_ws, size_t ws_size,
                              hipStream_t stream) {
  const float* inputs = (const float*)d_in[0];
  const float* evals  = (const float*)d_in[1];
  const float* W      = (const float*)d_in[2];
  const float* bias   = (const float*)d_in[3];
  const int*   esrc   = (const int*)d_in[4];
  const int*   edst   = (const int*)d_in[5];
  float* out = (float*)d_out;

  const size_t SZ = (size_t)NN * FF;        // floats per diffusion state
  float* B0 = (float*)d_ws;
  float* B1 = B0 + SZ;
  float* B2 = B1 + SZ;

  const dim3 blk(256);
  const int gridX0   = (NN * FF) / 256;     // 10000, exact
  const int gridNeg  = (NN * FF / 4) / 256; // 2500, exact
  const int gridScat = EE / 8;              // 80000, exact
  const int gridGemm = 20000 / 8;           // 2500, exact

  // m=0: x0 itself
  build_x0<<<gridX0, blk, 0, stream>>>(inputs, B0);
  gemm_acc<<<gridGemm, blk, 0, stream>>>(B0, W, bias, out, 0, 0);

  // ---- support 0 ----
  hipMemsetAsync(B1, 0, SZ * sizeof(float), stream);
  spmm_scatter<<<gridScat, blk, 0, stream>>>(evals, esrc, edst, B0, B1, 1.0f);
  gemm_acc<<<gridGemm, blk, 0, stream>>>(B1, W, bias, out, 1, 1);      // x1 = A0 x0

  neg_copy<<<gridNeg, blk, 0, stream>>>(B2, B0);
  spmm_scatter<<<gridScat, blk, 0, stream>>>(evals, esrc, edst, B1, B2, 2.0f);
  gemm_acc<<<gridGemm, blk, 0, stream>>>(B2, W, bias, out, 2, 1);      // T2 = 2A0x1 - x0

  neg_copy<<<gridNeg, blk, 0, stream>>>(B0, B1);
  spmm_scatter<<<gridScat, blk, 0, stream>>>(evals, esrc, edst, B2, B0, 2.0f);
  gemm_acc<<<gridGemm, blk, 0, stream>>>(B0, W, bias, out, 3, 1);      // T3 = 2A0T2 - x1

  // ---- support 1 (starts from mutated x0 = B2 = T2 of support 0) ----
  const float* v1 = evals + EE;
  const int*   s1 = esrc + EE;
  const int*   d1 = edst + EE;

  hipMemsetAsync(B1, 0, SZ * sizeof(float), stream);
  spmm_scatter<<<gridScat, blk, 0, stream>>>(v1, s1, d1, B2, B1, 1.0f);
  gemm_acc<<<gridGemm, blk, 0, stream>>>(B1, W, bias, out, 4, 1);      // x1' = A1 B2

  neg_copy<<<gridNeg, blk, 0, stream>>>(B0, B2);
  spmm_scatter<<<gridScat, blk, 0, stream>>>(v1, s1, d1, B1, B0, 2.0f);
  gemm_acc<<<gridGemm, blk, 0, stream>>>(B0, W, bias, out, 5, 1);      // 2A1x1' - B2

  neg_copy<<<gridNeg, blk, 0, stream>>>(B2, B1);
  spmm_scatter<<<gridScat, blk, 0, stream>>>(v1, s1, d1, B0, B2, 2.0f);
  gemm_acc<<<gridGemm, blk, 0, stream>>>(B2, W, bias, out, 6, 1);      // 2A1x2' - x1'
}